// SNN_FC_Layer_2405181686119
// MI455X (gfx1250) — compile-verified
//
#include <hip/hip_runtime.h>

typedef float v2f __attribute__((ext_vector_type(2)));
typedef float v8f __attribute__((ext_vector_type(8)));

#define MAXT 100000.0f
constexpr int Nn = 1024;   // inner dim (sorted/scan axis)
constexpr int Mm = 512;    // outputs per batch

// One block per batch row. 512 threads = 16 waves (wave32).
// Phase 1: bitonic argsort of x[b] in LDS.
// Phase 2: chunked (16-row) inclusive prefix sums via triangular-matrix WMMA,
//          elementwise spike-time test, min-reduce over the scan axis.
__launch_bounds__(512, 2)
__global__ void snn_fc_spike_kernel(const float* __restrict__ x,
                                    const float* __restrict__ w,
                                    float* __restrict__ out)
{
    __shared__ float s_xs[Nn];
    __shared__ int   s_idx[Nn];

    const int b   = blockIdx.x;
    const int tid = threadIdx.x;

    // ---- load + bitonic sort (ascending) of (x value, original index) ----
    for (int i = tid; i < Nn; i += 512) {
        s_xs[i]  = x[b * Nn + i];
        s_idx[i] = i;
    }
    __syncthreads();
    for (int k = 2; k <= Nn; k <<= 1) {
        for (int j = k >> 1; j > 0; j >>= 1) {
            for (int i = tid; i < Nn; i += 512) {
                int ixj = i ^ j;
                if (ixj > i) {
                    bool up = ((i & k) == 0);
                    float a = s_xs[i], c = s_xs[ixj];
                    if (up ? (a > c) : (a < c)) {
                        s_xs[i] = c; s_xs[ixj] = a;
                        int t = s_idx[i]; s_idx[i] = s_idx[ixj]; s_idx[ixj] = t;
                    }
                }
            }
            __syncthreads();
        }
    }

    // ---- phase 2: WMMA prefix-sum scan ----
    const int lane = tid & 31;
    const int ln   = lane & 15;   // column within 16x16 tile (= m offset, = A row n)
    const int h    = lane >> 4;   // lane half: selects K/row groups per ISA layouts
    const int wv   = tid >> 5;    // wave id 0..15
    const int m0   = wv * 32;     // this wave owns columns [m0, m0+32)

    // Constant lower-triangular ones matrix L, sliced into 16x4 A operands.
    // A layout (f32 16x4): lane holds row M=ln; VGPR0/1 = K = {4kk+2h, 4kk+2h+1}.
    v2f A1[4];
#pragma unroll
    for (int kk = 0; kk < 4; ++kk) {
        int j0 = 4 * kk + 2 * h;
        A1[kk][0] = (j0     <= ln) ? 1.0f : 0.0f;
        A1[kk][1] = (j0 + 1 <= ln) ? 1.0f : 0.0f;
    }

    float cw[2]  = {0.0f, 0.0f};   // carry of w_cum per tile (column m0+16t+ln)
    float cwi[2] = {0.0f, 0.0f};   // carry of wi_cum
    float mn[2]  = {MAXT, MAXT};   // running min over scan axis

    for (int c = 0; c < Nn / 16; ++c) {
        const int base = c * 16;

        // xs for the 8 scan rows this lane-half owns in the C layout (n = r + 8h),
        // plus the "next sorted value" for the window test.
        float xr[8], xnr[8];
#pragma unroll
        for (int r = 0; r < 8; ++r) xr[r] = s_xs[base + 8 * h + r];
#pragma unroll
        for (int r = 0; r < 7; ++r) xnr[r] = xr[r + 1];
        {
            int ng = base + 8 * h + 8;
            xnr[7] = (ng < Nn) ? s_xs[ng] : MAXT;
        }

        // A2 = L * diag(xs_chunk)  (for wi_cum), plus gather row indices for B.
        v2f A2[4];
        int r0[4], r1[4];
#pragma unroll
        for (int kk = 0; kk < 4; ++kk) {
            int j0 = 4 * kk + 2 * h;
            float xj0 = s_xs[base + j0];
            float xj1 = s_xs[base + j0 + 1];
            A2[kk][0] = (j0     <= ln) ? xj0 : 0.0f;
            A2[kk][1] = (j0 + 1 <= ln) ? xj1 : 0.0f;
            r0[kk] = s_idx[base + j0];
            r1[kk] = s_idx[base + j0 + 1];
        }

        // Warm L2/L0 for next chunk's gathered rows (global_prefetch_b8).
        if (c + 1 < Nn / 16) {
#pragma unroll
            for (int kk = 0; kk < 4; ++kk) {
                int nj = base + 16 + 4 * kk + 2 * h;
                __builtin_prefetch(&w[(size_t)s_idx[nj] * Mm + m0 + ln], 0, 3);
                __builtin_prefetch(&w[(size_t)s_idx[nj + 1] * Mm + m0 + ln], 0, 3);
            }
        }

#pragma unroll
        for (int t = 0; t < 2; ++t) {
            const int m = m0 + 16 * t + ln;

            // C initialized to per-column carry (all 8 rows identical).
            v8f accW, accWI;
#pragma unroll
            for (int e = 0; e < 8; ++e) { accW[e] = cw[t]; accWI[e] = cwi[t]; }

            // D = L x Wchunk + C  and  D = (L diag(xs)) x Wchunk + C,
            // chained over K in 4 steps of V_WMMA_F32_16X16X4_F32.
#pragma unroll
            for (int kk = 0; kk < 4; ++kk) {
                v2f bv;
                bv[0] = w[(size_t)r0[kk] * Mm + m];   // B: lane = column, K = j0
                bv[1] = w[(size_t)r1[kk] * Mm + m];   //    K = j0+1
                accW  = __builtin_amdgcn_wmma_f32_16x16x4_f32(
                            false, A1[kk], false, bv, (short)0, accW,  false, false);
                accWI = __builtin_amdgcn_wmma_f32_16x16x4_f32(
                            false, A2[kk], false, bv, (short)0, accWI, false, false);
            }

            // Elementwise spike-time computation on all 16 per-step values.
            // Fast reciprocal (v_rcp_f32, ~1 ulp) instead of full IEEE divide:
            // the huge-value cases (den clamped to 1e-10) are filtered to MAXT
            // by the window test, so rcp accuracy is ample here.
            float o8[8];
#pragma unroll
            for (int r = 0; r < 8; ++r) {
                float wc  = accW[r];
                float wic = accWI[r];
                float den = fminf(fmaxf(wc - 1.0f, 1e-10f), 1e10f);
                float o   = wic * __builtin_amdgcn_rcpf(den);
                bool bad  = (wc < 1.0f) | (o < xr[r]) | (o > xnr[r]);
                o8[r] = bad ? MAXT : o;
            }
            // log-depth min tree (lowers to v_min3_num_f32)
            float m01 = fminf(o8[0], o8[1]);
            float m23 = fminf(o8[2], o8[3]);
            float m45 = fminf(o8[4], o8[5]);
            float m67 = fminf(o8[6], o8[7]);
            mn[t] = fminf(mn[t], fminf(fminf(m01, m23), fminf(m45, m67)));

            // Carry = cumulative value at chunk row 15: VGPR 7 of lanes 16..31.
            cw[t]  = __shfl(accW[7],  ln + 16, 32);
            cwi[t] = __shfl(accWI[7], ln + 16, 32);
        }
    }

    // Combine the two lane-halves (rows 0-7 vs 8-15 of every chunk) and store.
#pragma unroll
    for (int t = 0; t < 2; ++t) {
        float v = fminf(mn[t], __shfl_xor(mn[t], 16, 32));
        if (h == 0) out[(size_t)b * Mm + m0 + 16 * t + ln] = v;
    }
}

extern "C" void kernel_launch(void* const* d_in, const int* in_sizes, int n_in,
                              void* d_out, int out_size, void* d_ws, size_t ws_size,
                              hipStream_t stream)
{
    (void)n_in; (void)out_size; (void)d_ws; (void)ws_size;
    const float* x = (const float*)d_in[0];   // (B, 1024) fp32
    const float* w = (const float*)d_in[1];   // (1024, 512) fp32
    float* out = (float*)d_out;               // (B, 512) fp32
    const int B = in_sizes[0] / Nn;
    snn_fc_spike_kernel<<<dim3(B), dim3(512), 0, stream>>>(x, w, out);
}